// SignCLLoss_49228915147547
// MI455X (gfx1250) — compile-verified
//
#include <hip/hip_runtime.h>

#define B_    64
#define T_    512
#define D_    1024
#define K_    10
#define TA_   510                      // number of anchor positions (t = 1..510)
#define INVT  14.2857142857142857f     // 1 / 0.07
#define EPS_  1e-8f

typedef float v2f __attribute__((ext_vector_type(2)));
typedef float v8f __attribute__((ext_vector_type(8)));

// ---------------------------------------------------------------------------
// Pass 1: reciprocal L2 norms for every feature row (B*T rows), one block/row.
// Also zero-initializes the scalar output (stream order guarantees it happens
// before the main kernel's atomics).
// ---------------------------------------------------------------------------
__global__ void __launch_bounds__(256)
rnorm_kernel(const float* __restrict__ F, float* __restrict__ rnorm,
             float* __restrict__ out) {
    const int row  = blockIdx.x;        // 0 .. B*T-1
    const int tid  = threadIdx.x;       // 0 .. 255
    if (row == 0 && tid == 0) out[0] = 0.0f;

    const float4 v = *(const float4*)(F + (size_t)row * D_ + (size_t)tid * 4);
    float ss = v.x * v.x + v.y * v.y + v.z * v.z + v.w * v.w;
    #pragma unroll
    for (int off = 1; off < 32; off <<= 1) ss += __shfl_xor(ss, off, 32);

    __shared__ float red[8];
    const int wave = tid >> 5, lane = tid & 31;
    if (lane == 0) red[wave] = ss;
    __syncthreads();
    if (tid == 0) {
        float t = 0.f;
        #pragma unroll
        for (int i = 0; i < 8; ++i) t += red[i];
        rnorm[row] = 1.0f / fmaxf(sqrtf(t), EPS_);
    }
}

// ---------------------------------------------------------------------------
// Pass 2: gather + normalize negatives into WMMA-B-friendly layout
// negsT[b][d][n], n padded 10 -> 16 with zeros. Block = (b, n).
// ---------------------------------------------------------------------------
__global__ void __launch_bounds__(256)
negs_kernel(const float* __restrict__ F, const long long* __restrict__ nbv,
            const long long* __restrict__ ntv, float* __restrict__ negsT) {
    const int b = blockIdx.x, n = blockIdx.y, tid = threadIdx.x;
    float* dst = negsT + (size_t)b * D_ * 16;

    if (n >= K_) {                      // zero padding columns 10..15
        #pragma unroll
        for (int j = 0; j < 4; ++j)
            dst[(size_t)(tid * 4 + j) * 16 + n] = 0.0f;
        return;
    }

    const long long rb = nbv[b * K_ + n];
    const long long rt = ntv[b * K_ + n];
    const float* src = F + ((size_t)rb * T_ + (size_t)rt) * D_;
    const float4 v = *(const float4*)(src + (size_t)tid * 4);

    float ss = v.x * v.x + v.y * v.y + v.z * v.z + v.w * v.w;
    #pragma unroll
    for (int off = 1; off < 32; off <<= 1) ss += __shfl_xor(ss, off, 32);

    __shared__ float red[8];
    __shared__ float rn_s;
    const int wave = tid >> 5, lane = tid & 31;
    if (lane == 0) red[wave] = ss;
    __syncthreads();
    if (tid == 0) {
        float t = 0.f;
        #pragma unroll
        for (int i = 0; i < 8; ++i) t += red[i];
        rn_s = 1.0f / fmaxf(sqrtf(t), EPS_);
    }
    __syncthreads();
    const float rn = rn_s;

    dst[(size_t)(tid * 4 + 0) * 16 + n] = v.x * rn;
    dst[(size_t)(tid * 4 + 1) * 16 + n] = v.y * rn;
    dst[(size_t)(tid * 4 + 2) * 16 + n] = v.z * rn;
    dst[(size_t)(tid * 4 + 3) * 16 + n] = v.w * rn;
}

// ---------------------------------------------------------------------------
// Pass 3: main loss kernel. Grid = (B, 4), block = 256 (8 waves).
// Each wave owns a 16-anchor tile: V_WMMA_F32_16X16X4_F32 accumulates the
// 16x16 anchor x negative similarity tile over D=1024; the positive dot
// product rides in the same loop on the VALU.
// ---------------------------------------------------------------------------
__global__ void __launch_bounds__(256)
closs_kernel(const float* __restrict__ F, const float* __restrict__ rnorm,
             const float* __restrict__ negsT, float* __restrict__ out) {
    const int b    = blockIdx.x;              // batch
    const int base = blockIdx.y * 128 + (threadIdx.x >> 5) * 16; // tile base anchor
    const int lane = threadIdx.x & 31;
    const int half = lane >> 4;               // 0: lanes 0-15, 1: lanes 16-31
    const int nm   = lane & 15;               // N index (B/C frag) == M index (A frag)
    const int koff = half * 2;                // k = koff + v  (ISA 16x4 fp32 layout)

    const int mrow   = base + nm;
    const int mclamp = (mrow < TA_) ? mrow : (TA_ - 1);
    const int ta     = 1 + mclamp;            // anchor row; positive row = ta+1
    const float* aptr = F + ((size_t)b * T_ + ta) * D_;
    const float* pptr = aptr + D_;
    const float  ra   = rnorm[b * T_ + ta];
    const float  rp   = rnorm[b * T_ + ta + 1];
    const float* nbase = negsT + (size_t)b * D_ * 16;

    v8f  c   = {0.f, 0.f, 0.f, 0.f, 0.f, 0.f, 0.f, 0.f};
    float acc = 0.f;                          // partial anchor·positive (ra folded in)

    #pragma unroll 4
    for (int d = 0; d < D_; d += 4) {
        const float2 av = *(const float2*)(aptr + d + koff);
        const float2 pv = *(const float2*)(pptr + d + koff);
        v2f a;  a.x  = av.x * ra;  a.y = av.y * ra;          // A fragment (normalized)
        v2f bb; bb.x = nbase[(d + koff) * 16 + nm];          // B fragment (pre-normalized)
                bb.y = nbase[(d + koff + 1) * 16 + nm];
        c = __builtin_amdgcn_wmma_f32_16x16x4_f32(
                /*neg_a=*/false, a, /*neg_b=*/false, bb,
                /*c_mod=*/(short)0, c, /*reuse_a=*/false, /*reuse_b=*/false);
        acc = fmaf(a.x, pv.x, acc);
        acc = fmaf(a.y, pv.y, acc);
    }

    // Full positive similarity for row (base+nm): combine the two K-halves.
    acc += __shfl_xor(acc, 16, 32);
    const float pos_sim = acc * rp * INVT;

    // Sum exp(sim/T) over the 10 valid negative columns within each 16-lane half.
    float s[8];
    #pragma unroll
    for (int v = 0; v < 8; ++v) {
        float e = (nm < K_) ? __expf(c[v] * INVT) : 0.f;
        e += __shfl_xor(e, 1, 32);
        e += __shfl_xor(e, 2, 32);
        e += __shfl_xor(e, 4, 32);
        e += __shfl_xor(e, 8, 32);
        s[v] = e;                 // valid for rows  m = v + 8*half  of this tile
    }

    // Owner lanes: 0..7 -> rows base+0..7 ; 24..31 -> rows base+8..15.
    // Those lanes already hold pos_sim for exactly that row.
    const int  vsel = lane & 7;
    float en = s[0];
    if (vsel == 1) en = s[1];
    if (vsel == 2) en = s[2];
    if (vsel == 3) en = s[3];
    if (vsel == 4) en = s[4];
    if (vsel == 5) en = s[5];
    if (vsel == 6) en = s[6];
    if (vsel == 7) en = s[7];

    const bool owner = (lane < 8) || (lane >= 24);
    const int  myrow = base + half * 8 + vsel;
    float lossv = 0.f;
    if (owner && myrow < TA_)
        lossv = __logf(1.0f + en * __expf(-pos_sim));  // -log(ep/(ep+en))

    lossv += __shfl_xor(lossv, 1, 32);
    lossv += __shfl_xor(lossv, 2, 32);
    lossv += __shfl_xor(lossv, 4, 32);
    lossv += __shfl_xor(lossv, 8, 32);
    lossv += __shfl_xor(lossv, 16, 32);
    if (lane == 0)
        atomicAdd(out, lossv * (1.0f / (float)(B_ * TA_)));
}

// ---------------------------------------------------------------------------
extern "C" void kernel_launch(void* const* d_in, const int* in_sizes, int n_in,
                              void* d_out, int out_size, void* d_ws, size_t ws_size,
                              hipStream_t stream) {
    (void)in_sizes; (void)n_in; (void)out_size; (void)ws_size;
    const float*     F   = (const float*)d_in[0];
    const long long* nbv = (const long long*)d_in[1];
    const long long* ntv = (const long long*)d_in[2];
    float* out = (float*)d_out;

    float* rnorm = (float*)d_ws;               // B*T floats        (128 KB)
    float* negsT = rnorm + (size_t)B_ * T_;    // B*D*16 floats     (4 MB)

    rnorm_kernel<<<B_ * T_, 256, 0, stream>>>(F, rnorm, out);
    negs_kernel<<<dim3(B_, 16), 256, 0, stream>>>(F, nbv, ntv, negsT);
    closs_kernel<<<dim3(B_, 4), 256, 0, stream>>>(F, rnorm, negsT, out);
}